// SinglePCAWrapper_84138409329079
// MI455X (gfx1250) — compile-verified
//
#include <hip/hip_runtime.h>
#include <math.h>

namespace {
constexpr int kBatches        = 2048;
constexpr int kPts            = 8192;
constexpr int kFloatsPerBatch = kPts * 3;              // 24576 floats = 6144 float4
constexpr int kThreads        = 256;
constexpr int kIters          = (kPts / 4) / kThreads; // 8 groups of 4 points per thread
}

typedef float v2f __attribute__((ext_vector_type(2)));
typedef float v8f __attribute__((ext_vector_type(8)));

__device__ __forceinline__ void cross3(float ax, float ay, float az,
                                       float bx, float by, float bz,
                                       float& cx, float& cy, float& cz) {
  cx = ay * bz - az * by;
  cy = az * bx - ax * bz;
  cz = ax * by - ay * bx;
}

// Eigenvector of symmetric 3x3 A for eigenvalue lam: best cross product of rows of (A - lam*I).
__device__ __forceinline__ void eigvec3(float a00, float a01, float a02,
                                        float a11, float a12, float a22,
                                        float lam,
                                        float& vx, float& vy, float& vz) {
  const float r0x = a00 - lam, r0y = a01,       r0z = a02;
  const float r1x = a01,       r1y = a11 - lam, r1z = a12;
  const float r2x = a02,       r2y = a12,       r2z = a22 - lam;
  float c0x, c0y, c0z, c1x, c1y, c1z, c2x, c2y, c2z;
  cross3(r0x, r0y, r0z, r1x, r1y, r1z, c0x, c0y, c0z);
  cross3(r0x, r0y, r0z, r2x, r2y, r2z, c1x, c1y, c1z);
  cross3(r1x, r1y, r1z, r2x, r2y, r2z, c2x, c2y, c2z);
  const float n0 = c0x * c0x + c0y * c0y + c0z * c0z;
  const float n1 = c1x * c1x + c1y * c1y + c1z * c1z;
  const float n2 = c2x * c2x + c2y * c2y + c2z * c2z;
  float bx = c0x, by = c0y, bz = c0z, bn = n0;
  if (n1 > bn) { bx = c1x; by = c1y; bz = c1z; bn = n1; }
  if (n2 > bn) { bx = c2x; by = c2y; bz = c2z; bn = n2; }
  const float inv = rsqrtf(fmaxf(bn, 1e-30f));
  vx = bx * inv; vy = by * inv; vz = bz * inv;
}

__global__ __launch_bounds__(kThreads)
void pca_fused(const float* __restrict__ x, float* __restrict__ out) {
  const int b   = blockIdx.x;
  const int tid = threadIdx.x;

  const float4* __restrict__ xb =
      reinterpret_cast<const float4*>(x + (size_t)b * kFloatsPerBatch);
  float4* __restrict__ ob =
      reinterpret_cast<float4*>(out + (size_t)b * kFloatsPerBatch);

  // ---- Pass A: load 32 points (24 float4, all B128/coalesced), accumulate moments.
  float4 h[kIters][3];
  float s0 = 0.f, s1 = 0.f, s2 = 0.f;
  float m00 = 0.f, m01 = 0.f, m02 = 0.f, m11 = 0.f, m12 = 0.f, m22 = 0.f;

  #pragma unroll
  for (int i = 0; i < kIters; ++i) {
    const int g = tid + i * kThreads;    // group of 4 points = 3 float4
    h[i][0] = xb[3 * g + 0];
    h[i][1] = xb[3 * g + 1];
    h[i][2] = xb[3 * g + 2];
  }

  auto acc = [&](float px, float py, float pz) {
    s0 += px; s1 += py; s2 += pz;
    m00 = fmaf(px, px, m00); m01 = fmaf(px, py, m01); m02 = fmaf(px, pz, m02);
    m11 = fmaf(py, py, m11); m12 = fmaf(py, pz, m12); m22 = fmaf(pz, pz, m22);
  };

  #pragma unroll
  for (int i = 0; i < kIters; ++i) {
    const float4 A = h[i][0], B4 = h[i][1], C4 = h[i][2];
    acc(A.x,  A.y,  A.z);
    acc(A.w,  B4.x, B4.y);
    acc(B4.z, B4.w, C4.x);
    acc(C4.y, C4.z, C4.w);
  }

  // ---- Block reduction of the 9 partial sums using V_WMMA_F32_16X16X4_F32.
  // LDS matrix: 256 rows x 16 cols (cols 0..8 = partials). A = e0-row of ones,
  // so each WMMA adds 4 rows of B into D row 0; 64 iterations reduce all 256 rows.
  __shared__ float red[kThreads * 16];
  __shared__ float tot[16];
  __shared__ float sp[12];

  float* r = &red[tid * 16];
  r[0] = s0;  r[1] = s1;  r[2] = s2;
  r[3] = m00; r[4] = m01; r[5] = m02; r[6] = m11; r[7] = m12; r[8] = m22;
  #pragma unroll
  for (int j = 9; j < 16; ++j) r[j] = 0.f;
  __syncthreads();

  if (tid < 32) {                       // whole wave 0 -> EXEC all ones for WMMA
    v2f a;
    const float one = ((tid & 15) == 0) ? 1.0f : 0.0f;  // A[0][k]=1 (lane0: K0,K1; lane16: K2,K3)
    a[0] = one; a[1] = one;
    v8f c = {0.f, 0.f, 0.f, 0.f, 0.f, 0.f, 0.f, 0.f};
    const int col  = tid & 15;
    const int hrow = (tid >> 4) << 1;   // lanes 0-15: K rows +0/+1, lanes 16-31: +2/+3
    for (int ch = 0; ch < kThreads / 4; ++ch) {
      const int r0 = ch * 4 + hrow;
      v2f bb;
      bb[0] = red[(r0    ) * 16 + col];
      bb[1] = red[(r0 + 1) * 16 + col];
      c = __builtin_amdgcn_wmma_f32_16x16x4_f32(false, a, false, bb,
                                                (short)0, c, false, false);
    }
    if (tid < 9) tot[tid] = c[0];       // D row 0 (VGPR0, lanes 0-15): column totals
  }
  __syncthreads();

  // ---- One thread: covariance + analytic 3x3 eigh (ascending, like jnp.linalg.eigh).
  if (tid == 0) {
    const float invN = 1.0f / (float)kPts;
    const float mu0 = tot[0] * invN, mu1 = tot[1] * invN, mu2 = tot[2] * invN;
    const float a00 = fmaf(-mu0, mu0, tot[3] * invN);
    const float a01 = fmaf(-mu0, mu1, tot[4] * invN);
    const float a02 = fmaf(-mu0, mu2, tot[5] * invN);
    const float a11 = fmaf(-mu1, mu1, tot[6] * invN);
    const float a12 = fmaf(-mu1, mu2, tot[7] * invN);
    const float a22 = fmaf(-mu2, mu2, tot[8] * invN);

    float v0x = 1.f, v0y = 0.f, v0z = 0.f;   // eigenvector columns (ascending)
    float v1x = 0.f, v1y = 1.f, v1z = 0.f;
    float v2x = 0.f, v2y = 0.f, v2z = 1.f;

    const float q   = (a00 + a11 + a22) * (1.0f / 3.0f);
    const float p1  = a01 * a01 + a02 * a02 + a12 * a12;
    const float b00 = a00 - q, b11 = a11 - q, b22 = a22 - q;
    const float p2  = b00 * b00 + b11 * b11 + b22 * b22 + 2.0f * p1;
    if (p2 > 1e-28f) {
      const float p  = sqrtf(p2 * (1.0f / 6.0f));
      const float ip = 1.0f / p;
      const float c00 = b00 * ip, c01 = a01 * ip, c02 = a02 * ip;
      const float c11 = b11 * ip, c12 = a12 * ip, c22 = b22 * ip;
      float detB = c00 * (c11 * c22 - c12 * c12)
                 - c01 * (c01 * c22 - c12 * c02)
                 + c02 * (c01 * c12 - c11 * c02);
      float rr = fminf(fmaxf(0.5f * detB, -1.0f), 1.0f);
      const float phi  = acosf(rr) * (1.0f / 3.0f);
      const float lBig = q + 2.0f * p * cosf(phi);                        // largest
      const float lSml = q + 2.0f * p * cosf(phi + 2.0943951023931953f);  // smallest
      eigvec3(a00, a01, a02, a11, a12, a22, lSml, v0x, v0y, v0z);
      eigvec3(a00, a01, a02, a11, a12, a22, lBig, v2x, v2y, v2z);
      cross3(v2x, v2y, v2z, v0x, v0y, v0z, v1x, v1y, v1z);               // middle = big x small
      const float nrm = rsqrtf(fmaxf(v1x * v1x + v1y * v1y + v1z * v1z, 1e-30f));
      v1x *= nrm; v1y *= nrm; v1z *= nrm;
    }
    sp[0] = mu0; sp[1] = mu1; sp[2] = mu2;
    sp[3] = v0x; sp[4]  = v1x; sp[5]  = v2x;   // E[d][e] row-major, columns = eigenvectors
    sp[6] = v0y; sp[7]  = v1y; sp[8]  = v2y;
    sp[9] = v0z; sp[10] = v1z; sp[11] = v2z;
  }
  __syncthreads();

  // ---- Pass B: project register-resident points, store coalesced B128.
  const float mu0 = sp[0], mu1 = sp[1], mu2 = sp[2];
  const float e00 = sp[3], e01 = sp[4],  e02 = sp[5];
  const float e10 = sp[6], e11 = sp[7],  e12 = sp[8];
  const float e20 = sp[9], e21 = sp[10], e22 = sp[11];

  auto proj = [&](float px, float py, float pz, float& ox, float& oy, float& oz) {
    const float cx = px - mu0, cy = py - mu1, cz = pz - mu2;
    ox = fmaf(cx, e00, fmaf(cy, e10, cz * e20));
    oy = fmaf(cx, e01, fmaf(cy, e11, cz * e21));
    oz = fmaf(cx, e02, fmaf(cy, e12, cz * e22));
  };

  #pragma unroll
  for (int i = 0; i < kIters; ++i) {
    const int g = tid + i * kThreads;
    const float4 A = h[i][0], B4 = h[i][1], C4 = h[i][2];
    float o[12];
    proj(A.x,  A.y,  A.z,  o[0], o[1],  o[2]);
    proj(A.w,  B4.x, B4.y, o[3], o[4],  o[5]);
    proj(B4.z, B4.w, C4.x, o[6], o[7],  o[8]);
    proj(C4.y, C4.z, C4.w, o[9], o[10], o[11]);
    ob[3 * g + 0] = make_float4(o[0], o[1], o[2],  o[3]);
    ob[3 * g + 1] = make_float4(o[4], o[5], o[6],  o[7]);
    ob[3 * g + 2] = make_float4(o[8], o[9], o[10], o[11]);
  }
}

extern "C" void kernel_launch(void* const* d_in, const int* in_sizes, int n_in,
                              void* d_out, int out_size, void* d_ws, size_t ws_size,
                              hipStream_t stream) {
  (void)in_sizes; (void)n_in; (void)out_size; (void)d_ws; (void)ws_size;
  const float* x = (const float*)d_in[0];
  float* out = (float*)d_out;
  pca_fused<<<dim3(kBatches), dim3(kThreads), 0, stream>>>(x, out);
}